// DetectionSampler_46385646797219
// MI455X (gfx1250) — compile-verified
//
#include <hip/hip_runtime.h>
#include <hip/hip_bf16.h>

// ---------------- problem constants (from reference) ----------------
#define BATCH   8
#define DDIM    128
#define HH      256
#define WW      256
#define CELL    8
#define BORD    16
#define HC      28                 // (256-32)/8
#define WC      28
#define NCELLS  (HC*WC)            // 784
#define NPTS    (BATCH*NCELLS)     // 6272
#define NNEG    12                 // ring offsets (49<=r2<=64, step2)
#define SCW     (1 + NNEG + NPTS)  // 6285 score columns
#define PLANE   (HH*WW)            // 65536

typedef float v2f __attribute__((ext_vector_type(2)));
typedef float v8f __attribute__((ext_vector_type(8)));

__device__ __forceinline__ int iclamp(int v, int lo, int hi) {
    return v < lo ? lo : (v > hi ? hi : v);
}

__device__ __forceinline__ float wave_sum(float v) {
#pragma unroll
    for (int m = 16; m >= 1; m >>= 1) v += __shfl_xor(v, m, 32);
    return v;
}

// ---------------- kernel 1: cell argmax on det1 + flow sample ----------------
__global__ void k_sample1(const float* __restrict__ det1,
                          const float* __restrict__ aflow,
                          int* __restrict__ x1, int* __restrict__ y1,
                          int* __restrict__ x2, int* __restrict__ y2,
                          float* __restrict__ mask_out) {
    int n = blockIdx.x * blockDim.x + threadIdx.x;
    if (n >= NPTS) return;
    int b = n / NCELLS, rem = n % NCELLS;
    int cy = rem / WC, cx = rem % WC;
    const float* dp = det1 + (size_t)b * PLANE;
    int y0 = BORD + cy * CELL, x0 = BORD + cx * CELL;
    float best = -INFINITY; int bi = 0, bj = 0;
#pragma unroll
    for (int ii = 0; ii < CELL; ++ii)
#pragma unroll
        for (int jj = 0; jj < CELL; ++jj) {
            float v = dp[(y0 + ii) * WW + (x0 + jj)];
            if (v > best) { best = v; bi = ii; bj = jj; }   // first max wins
        }
    int yy = y0 + bi, xx = x0 + bj;
    x1[n] = xx; y1[n] = yy;
    const float* fl = aflow + (size_t)b * 2 * PLANE;
    float fx = fl[yy * WW + xx];
    float fy = fl[PLANE + yy * WW + xx];
    int xi = (int)(fx + 0.5f);   // trunc toward zero, matches astype(int32)
    int yi = (int)(fy + 0.5f);
    x2[n] = xi; y2[n] = yi;
    mask_out[n] = (xi >= 0 && yi >= 0 && xi < WW && yi < HH) ? 1.0f : 0.0f;
}

// ---------------- kernel 2: cell argmax on det2 (distractors) ----------------
__global__ void k_sample2(const float* __restrict__ det2,
                          int* __restrict__ xd, int* __restrict__ yd) {
    int n = blockIdx.x * blockDim.x + threadIdx.x;
    if (n >= NPTS) return;
    int b = n / NCELLS, rem = n % NCELLS;
    int cy = rem / WC, cx = rem % WC;
    const float* dp = det2 + (size_t)b * PLANE;
    int y0 = BORD + cy * CELL, x0 = BORD + cx * CELL;
    float best = -INFINITY; int bi = 0, bj = 0;
#pragma unroll
    for (int ii = 0; ii < CELL; ++ii)
#pragma unroll
        for (int jj = 0; jj < CELL; ++jj) {
            float v = dp[(y0 + ii) * WW + (x0 + jj)];
            if (v > best) { best = v; bi = ii; bj = jj; }
        }
    xd[n] = x0 + bj; yd[n] = y0 + bi;
}

// ---------------- kernel 3: gather descriptor rows (channel-strided -> dense) ----------------
__global__ void k_gather(const float* __restrict__ des,
                         const int* __restrict__ xs, const int* __restrict__ ys,
                         float* __restrict__ rows) {
    int wave = (blockIdx.x * blockDim.x + threadIdx.x) >> 5;
    int lane = threadIdx.x & 31;
    if (wave >= NPTS) return;
    int b = wave / NCELLS;
    int x = xs[wave], y = ys[wave];
    int c0 = lane * 4;
    const float* p = des + ((size_t)b * DDIM + c0) * PLANE + (size_t)y * WW + x;
    float4 v;
    v.x = p[0]; v.y = p[PLANE]; v.z = p[2 * PLANE]; v.w = p[3 * PLANE];
    *(float4*)(rows + (size_t)wave * DDIM + c0) = v;
}

// ---------------- kernel 4: positive / negative offset scores + qlt ----------------
__global__ void k_posneg(const float* __restrict__ sdes1,
                         const float* __restrict__ des2,
                         const float* __restrict__ qlt1,
                         const float* __restrict__ qlt2,
                         const int* __restrict__ x1, const int* __restrict__ y1,
                         const int* __restrict__ x2, const int* __restrict__ y2,
                         float* __restrict__ scores, float* __restrict__ qlt_out) {
    int wave = (blockIdx.x * blockDim.x + threadIdx.x) >> 5;
    int lane = threadIdx.x & 31;
    if (wave >= NPTS) return;
    int b = wave / NCELLS;
    int c0 = lane * 4;
    const float4 s = *(const float4*)(sdes1 + (size_t)wave * DDIM + c0);
    const float* dbase = des2 + ((size_t)b * DDIM + c0) * PLANE;
    int xi = x2[wave], yi = y2[wave];

    // positives: disk i*i+j*j <= 9, enumeration order == numpy comprehension
    float best = -INFINITY; int bdx = 0, bdy = 0;
    for (int i = -3; i <= 3; ++i)
        for (int j = -3; j <= 3; ++j) {
            if (i * i + j * j > 9) continue;
            int xc = iclamp(xi + i, 0, WW - 1);
            int yc = iclamp(yi + j, 0, HH - 1);
            const float* p = dbase + (size_t)yc * WW + xc;
            float d = s.x * p[0] + s.y * p[PLANE] + s.z * p[2 * PLANE] + s.w * p[3 * PLANE];
            d = wave_sum(d);                       // all lanes get total
            if (d > best) { best = d; bdx = i; bdy = j; }
        }
    if (lane == 0) {
        scores[(size_t)wave * SCW + 0] = best;
        int sx = iclamp(xi + bdx, 0, WW - 1);
        int sy = iclamp(yi + bdy, 0, HH - 1);
        float q1 = qlt1[(size_t)b * PLANE + (size_t)y1[wave] * WW + x1[wave]];
        float q2 = qlt2[(size_t)b * PLANE + (size_t)sy * WW + sx];
        qlt_out[wave] = 0.5f * (q1 + q2);
    }

    // negatives: ring 49 <= i*i+j*j <= 64, step 2
    int q = 0;
    for (int i = -8; i <= 8; i += 2)
        for (int j = -8; j <= 8; j += 2) {
            int r2 = i * i + j * j;
            if (r2 < 49 || r2 > 64) continue;
            int xc = iclamp(xi + i, 0, WW - 1);
            int yc = iclamp(yi + j, 0, HH - 1);
            const float* p = dbase + (size_t)yc * WW + xc;
            float d = s.x * p[0] + s.y * p[PLANE] + s.z * p[2 * PLANE] + s.w * p[3 * PLANE];
            d = wave_sum(d);
            if (lane == 0) scores[(size_t)wave * SCW + 1 + q] = d;
            ++q;
        }
}

// ---------------- kernel 5: labels fill ----------------
__global__ void k_labels(float* __restrict__ labels) {
    size_t idx = (size_t)blockIdx.x * blockDim.x + threadIdx.x;
    if (idx >= (size_t)NPTS * SCW) return;
    labels[idx] = (idx % SCW == 0) ? 1.0f : 0.0f;
}

// ---------------- kernel 6: distractor GEMM via V_WMMA_F32_16X16X4_F32 ----------------
// S (NPTS x 128) * T^T (128 x NPTS) -> scores[:, 13:13+NPTS], with distance/batch masking.
// Block: 128 threads = 4 waves; wave w owns rows [by*64 + 16w, +16), cols [bx*64, +64).
__global__ void __launch_bounds__(128) k_gemm(const float* __restrict__ S,
                                              const float* __restrict__ T,
                                              const int* __restrict__ x2,
                                              const int* __restrict__ y2,
                                              const int* __restrict__ xd,
                                              const int* __restrict__ yd,
                                              float* __restrict__ scores) {
    const int lane  = threadIdx.x & 31;
    const int wave  = threadIdx.x >> 5;
    const int ln    = lane & 15;        // M index for A rows, N index for B cols
    const int khalf = lane >> 4;        // 0: K={0,1}, 1: K={2,3} within a step
    const int m0 = blockIdx.y * 64 + wave * 16;
    const int n0 = blockIdx.x * 64;

    const float* arow = S + (size_t)(m0 + ln) * DDIM + khalf * 2;
    const float* brow0 = T + (size_t)(n0 + ln) * DDIM + khalf * 2;

    v8f acc0 = {}; v8f acc1 = {}; v8f acc2 = {}; v8f acc3 = {};
#pragma unroll 8
    for (int k = 0; k < DDIM; k += 4) {
        v2f a  = *(const v2f*)(arow + k);
        v2f b0 = *(const v2f*)(brow0 + k);
        v2f b1 = *(const v2f*)(brow0 + 16 * DDIM + k);
        v2f b2 = *(const v2f*)(brow0 + 32 * DDIM + k);
        v2f b3 = *(const v2f*)(brow0 + 48 * DDIM + k);
        acc0 = __builtin_amdgcn_wmma_f32_16x16x4_f32(false, a, false, b0, (short)0, acc0, false, false);
        acc1 = __builtin_amdgcn_wmma_f32_16x16x4_f32(false, a, false, b1, (short)0, acc1, false, false);
        acc2 = __builtin_amdgcn_wmma_f32_16x16x4_f32(false, a, false, b2, (short)0, acc2, false, false);
        acc3 = __builtin_amdgcn_wmma_f32_16x16x4_f32(false, a, false, b3, (short)0, acc3, false, false);
    }

    // mask + store: VGPR r of C/D holds (M = r + 8*khalf, N = ln)
#pragma unroll
    for (int r = 0; r < 8; ++r) {
        int gm = m0 + r + khalf * 8;
        int gmb = gm / NCELLS;
        int mx = x2[gm], my = y2[gm];
        float* orow = scores + (size_t)gm * SCW + 13;
#pragma unroll
        for (int t = 0; t < 4; ++t) {
            int gn = n0 + t * 16 + ln;
            float v = (t == 0) ? acc0[r] : (t == 1) ? acc1[r] : (t == 2) ? acc2[r] : acc3[r];
            int dx = xd[gn] - mx;
            int dy = yd[gn] - my;
            int dis2 = dx * dx + dy * dy + ((gn / NCELLS != gmb) ? 9 : 0);
            orow[gn] = (dis2 < 9) ? 0.0f : v;
        }
    }
}

// ---------------- launch ----------------
extern "C" void kernel_launch(void* const* d_in, const int* in_sizes, int n_in,
                              void* d_out, int out_size, void* d_ws, size_t ws_size,
                              hipStream_t stream) {
    const float* des1  = (const float*)d_in[0];
    const float* det1  = (const float*)d_in[1];
    const float* qlt1  = (const float*)d_in[2];
    const float* des2  = (const float*)d_in[3];
    const float* det2  = (const float*)d_in[4];
    const float* qlt2  = (const float*)d_in[5];
    const float* aflow = (const float*)d_in[6];

    // output layout: scores | labels | mask | qlt
    float* scores = (float*)d_out;
    float* labels = scores + (size_t)NPTS * SCW;
    float* mask_o = labels + (size_t)NPTS * SCW;
    float* qlt_o  = mask_o + NPTS;

    // workspace layout
    char* ws = (char*)d_ws;
    float* sdes1 = (float*)ws;                         ws += (size_t)NPTS * DDIM * 4;
    float* distr = (float*)ws;                         ws += (size_t)NPTS * DDIM * 4;
    int* x1 = (int*)ws; ws += NPTS * 4;
    int* y1 = (int*)ws; ws += NPTS * 4;
    int* x2 = (int*)ws; ws += NPTS * 4;
    int* y2 = (int*)ws; ws += NPTS * 4;
    int* xd = (int*)ws; ws += NPTS * 4;
    int* yd = (int*)ws; ws += NPTS * 4;

    dim3 b256(256);
    int nb = (NPTS + 255) / 256;
    k_sample1<<<nb, b256, 0, stream>>>(det1, aflow, x1, y1, x2, y2, mask_o);
    k_sample2<<<nb, b256, 0, stream>>>(det2, xd, yd);

    int gwaves = NPTS / 8;   // 8 waves per 256-thread block, 6272 % 8 == 0
    k_gather<<<gwaves, b256, 0, stream>>>(des1, x1, y1, sdes1);
    k_gather<<<gwaves, b256, 0, stream>>>(des2, xd, yd, distr);

    k_posneg<<<gwaves, b256, 0, stream>>>(sdes1, des2, qlt1, qlt2,
                                          x1, y1, x2, y2, scores, qlt_o);

    size_t ltot = (size_t)NPTS * SCW;
    k_labels<<<(unsigned)((ltot + 255) / 256), b256, 0, stream>>>(labels);

    dim3 ggrid(NPTS / 64, NPTS / 64);   // 98 x 98
    k_gemm<<<ggrid, dim3(128), 0, stream>>>(sdes1, distr, x2, y2, xd, yd, scores);
}